// ScaledDotProductAttention_4818953306885
// MI455X (gfx1250) — compile-verified
//
#include <hip/hip_runtime.h>
#include <hip/hip_bf16.h>

typedef __attribute__((ext_vector_type(2))) float v2f;
typedef __attribute__((ext_vector_type(4))) float v4f;
typedef __attribute__((ext_vector_type(8))) float v8f;

// Problem constants (from reference)
#define BATCH 64
#define SEQ   1024
#define HDIM  64

#define QT      16              // queries per workgroup (one WMMA M-tile)
#define WAVES   4               // wave32 x4 = 128 threads
#define THREADS (WAVES * 32)
#define SROW    (SEQ + 4)       // padded LDS row stride (floats); 1028*4 B = 16*257 -> 16B aligned rows

__global__ __launch_bounds__(THREADS)
void sdpa_f32wmma_kernel(const float* __restrict__ Q,
                         const float* __restrict__ K,
                         const float* __restrict__ V,
                         const float* __restrict__ scale_p,
                         float* __restrict__ ctx_out,
                         float* __restrict__ att_out)
{
    __shared__ float sS[QT * SROW];   // 16 x 1028 f32: scores, then unnormalized exp(S-m)
    __shared__ float sred[QT * 8];    // per-row partial reductions
    __shared__ float srow[2 * QT];    // [0..15] row max, [16..31] 1/rowsum

    const int tid  = threadIdx.x;
    const int lane = tid & 31;
    const int wave = tid >> 5;
    const int lrow = lane & 15;       // 0..15 : matrix row (A) / column (B)
    const int lhi  = lane >> 4;       // 0/1   : which half of the K-quad / row pair

    const int wg = blockIdx.x;        // BATCH * (SEQ/QT) = 4096 workgroups
    const int b  = wg >> 6;           // SEQ/QT == 64
    const int q0 = (wg & 63) * QT;

    const float scale = *scale_p;

    const float* Qb = Q + (size_t)b * SEQ * HDIM + (size_t)q0 * HDIM;
    const float* Kb = K + (size_t)b * SEQ * HDIM;
    const float* Vb = V + (size_t)b * SEQ * HDIM;

    // ---------------- preload A operands: Q tile 16x64, pre-scaled ----------------
    // A 16x4 layout: lane l holds row m=l%16; VGPR0/1 = dims {2*lhi, 2*lhi+1} of the quad.
    v2f aq[HDIM / 4];
    #pragma unroll
    for (int kk = 0; kk < HDIM / 4; ++kk) {
        const float* p = Qb + lrow * HDIM + kk * 4 + lhi * 2;
        v2f t = *(const v2f*)p;
        t.x *= scale; t.y *= scale;
        aq[kk] = t;
    }

    // ---------------- Phase 1: S = (Q*scale) . K^T -> LDS ----------------
    // Each wave computes 16 of the 64 16x16 key tiles; 16 chained fp32 WMMAs per tile.
    for (int t = wave; t < SEQ / 16; t += WAVES) {
        const int n0 = t * 16;
        v8f c = {};
        #pragma unroll
        for (int kk = 0; kk < HDIM / 4; ++kk) {
            // B 4x16 = K^T tile: lane l holds column n=l%16 (key n0+n),
            // VGPR0/1 = rows {2*lhi, 2*lhi+1} => contiguous dims of that key.
            const float* p = Kb + (size_t)(n0 + lrow) * HDIM + kk * 4 + lhi * 2;
            v2f bv = *(const v2f*)p;
            c = __builtin_amdgcn_wmma_f32_16x16x4_f32(false, aq[kk], false, bv,
                                                      (short)0, c, false, false);
        }
        // C/D layout: VGPR r, lane l -> row (r + 8*lhi), col l%16
        #pragma unroll
        for (int r = 0; r < 8; ++r) {
            const int m = r + 8 * lhi;
            sS[m * SROW + n0 + lrow] = c[r];
        }
    }
    __syncthreads();

    // ---------------- Phase 2: row max, then exp(S-m) in place + row sum ----------------
    {
        const int row = tid >> 3;     // 8 threads per row
        const int sub = tid & 7;
        float m = -3.402823466e+38f;
        for (int j = 0; j < SEQ / 32; ++j) {               // 32 float4 per thread, strided
            v4f x = *(const v4f*)&sS[row * SROW + (sub + j * 8) * 4];
            m = fmaxf(m, fmaxf(fmaxf(x.x, x.y), fmaxf(x.z, x.w)));
        }
        sred[row * 8 + sub] = m;
    }
    __syncthreads();
    if (tid < QT) {
        float m = sred[tid * 8];
        #pragma unroll
        for (int i = 1; i < 8; ++i) m = fmaxf(m, sred[tid * 8 + i]);
        srow[tid] = m;
    }
    __syncthreads();
    {
        const int row = tid >> 3;
        const int sub = tid & 7;
        const float m = srow[row];
        float s = 0.0f;
        for (int j = 0; j < SEQ / 32; ++j) {
            const int c = (sub + j * 8) * 4;
            v4f x = *(const v4f*)&sS[row * SROW + c];
            v4f p;
            p.x = __expf(x.x - m);
            p.y = __expf(x.y - m);
            p.z = __expf(x.z - m);
            p.w = __expf(x.w - m);
            s += (p.x + p.y) + (p.z + p.w);
            *(v4f*)&sS[row * SROW + c] = p;                // store UNNORMALIZED exp back
        }
        sred[row * 8 + sub] = s;
    }
    __syncthreads();
    if (tid < QT) {
        float s = 0.0f;
        #pragma unroll
        for (int i = 0; i < 8; ++i) s += sred[tid * 8 + i];
        srow[QT + tid] = 1.0f / s;
    }
    __syncthreads();
    // From here on sS is read-only (unnormalized probabilities): no more barriers needed.

    // ---------------- Phase 3a: stream normalized attention to global (b128 stores) ----------------
    // Issued first so the 268MB store stream drains while the PV WMMAs run below.
    float* attb = att_out + (size_t)b * SEQ * SEQ + (size_t)q0 * SEQ;
    for (int row = 0; row < QT; ++row) {
        const float inv = srow[QT + row];
        #pragma unroll
        for (int j = 0; j < SEQ / (THREADS * 4); ++j) {    // 2 float4 per thread per row
            const int c4 = tid + j * THREADS;              // lane-contiguous float4 index
            v4f p = *(const v4f*)&sS[row * SROW + 4 * c4];
            p *= inv;
            *(v4f*)&attb[(size_t)row * SEQ + 4 * c4] = p;
        }
    }

    // ---------------- Phase 3b: O = P_un . V, deferred normalization; wave w owns dims [16w,16w+16) ----
    v8f o = {};
    const int d0 = wave * 16;
    for (int t = 0; t < SEQ / 4; ++t) {                    // 4 keys per fp32 WMMA
        const int k0 = t * 4;
        // A = P 16x4: lane row lrow, keys {k0+2*lhi, +1}; 8B-aligned LDS b64 read,
        // row stride 1028 -> lanes land on distinct banks.
        v2f ap = *(const v2f*)&sS[lrow * SROW + k0 + lhi * 2];
        // B = V 4x16: lane column d0+lrow, rows (keys) {k0+2*lhi, +1}.
        v2f bv;
        bv.x = Vb[(size_t)(k0 + lhi * 2 + 0) * HDIM + d0 + lrow];
        bv.y = Vb[(size_t)(k0 + lhi * 2 + 1) * HDIM + d0 + lrow];
        o = __builtin_amdgcn_wmma_f32_16x16x4_f32(false, ap, false, bv,
                                                  (short)0, o, false, false);
    }
    float* ctxb = ctx_out + (size_t)b * SEQ * HDIM + (size_t)q0 * HDIM;
    #pragma unroll
    for (int r = 0; r < 8; ++r) {
        const int m = r + 8 * lhi;
        ctxb[(size_t)m * HDIM + d0 + lrow] = o[r] * srow[QT + m];   // normalize at the end
    }
}

extern "C" void kernel_launch(void* const* d_in, const int* in_sizes, int n_in,
                              void* d_out, int out_size, void* d_ws, size_t ws_size,
                              hipStream_t stream) {
    const float* q     = (const float*)d_in[0];
    const float* k     = (const float*)d_in[1];
    const float* v     = (const float*)d_in[2];
    const float* scale = (const float*)d_in[3];

    float* ctx = (float*)d_out;                                   // [B, L, D]
    float* att = ctx + (size_t)BATCH * SEQ * HDIM;                // [B, L, L]

    const int grid = BATCH * (SEQ / QT);                          // 4096 workgroups
    sdpa_f32wmma_kernel<<<grid, THREADS, 0, stream>>>(q, k, v, scale, ctx, att);
}